// RotH_781684048762
// MI455X (gfx1250) — compile-verified
//
#include <hip/hip_runtime.h>
#include <hip/hip_bf16.h>
#include <math.h>

typedef __attribute__((ext_vector_type(2))) float v2f;
typedef __attribute__((ext_vector_type(8))) float v8f;

#define DD   512
#define BB   128
#define MM   1024
#define MARGIN_F 9.0f
#define EPS_F    1e-6f
#define MINN_F   1e-15f

// ---------------------------------------------------------------------------
// Kernel A: transformed head  g[b][d] = rot(expmap0(head)+rc) - rc + rt + EPS
// One block per batch row b; 256 threads, one Givens pair (2 dims) per thread.
// ---------------------------------------------------------------------------
__global__ __launch_bounds__(256) void roth_head_kernel(
    const int* __restrict__ u_idx, const int* __restrict__ r_idx,
    const float* __restrict__ emb_entity,
    const float* __restrict__ relation_rot,
    const float* __restrict__ relation_rot_center,
    const float* __restrict__ relation_trans,
    float* __restrict__ g_ws)
{
    const int b = blockIdx.x;
    const int t = threadIdx.x;                 // pair index 0..255
    const int u = u_idx[b];
    const int r = r_idx[b];

    const float* hrow = emb_entity + (long long)u * DD;
    float h0 = hrow[2 * t];
    float h1 = hrow[2 * t + 1];

    // block reduction for ||head||
    __shared__ float red[256];
    red[t] = h0 * h0 + h1 * h1;
    __syncthreads();
    for (int s = 128; s > 0; s >>= 1) {
        if (t < s) red[t] += red[t + s];
        __syncthreads();
    }
    float n = fmaxf(sqrtf(red[0]), MINN_F);
    float factor = tanhf(n) / n;               // expmap0, c = 1

    const float* rc = relation_rot_center + (long long)r * DD;
    const float* rr = relation_rot        + (long long)r * DD;
    const float* rt = relation_trans      + (long long)r * DD;

    float c0 = rc[2 * t], c1 = rc[2 * t + 1];
    float x0 = h0 * factor + c0;
    float x1 = h1 * factor + c1;

    float g0 = rr[2 * t], g1 = rr[2 * t + 1];
    float gn = fmaxf(sqrtf(g0 * g0 + g1 * g1), MINN_F);
    g0 /= gn; g1 /= gn;

    float o0 = g0 * x0 - g1 * x1;
    float o1 = g0 * x1 + g1 * x0;

    o0 = o0 - c0 + rt[2 * t]     + EPS_F;      // fold +EPS into g
    o1 = o1 - c1 + rt[2 * t + 1] + EPS_F;

    g_ws[b * DD + 2 * t]     = o0;
    g_ws[b * DD + 2 * t + 1] = o1;
}

// ---------------------------------------------------------------------------
// Kernel B: scores via fp32 WMMA Gram-diagonal trick.
// Block = 4 waves, all on the same batch row b; each wave owns 16 tails.
// Per K-chunk of 4 dims: lane (m + 16c) holds diff for tail m, dims {k+2c, k+2c+1}
// == A-layout of V_WMMA_F32_16X16X4_F32; B = A^T has identical register image,
// so acc = A*A^T accumulates ||diff_i||^2 on the diagonal.
// ---------------------------------------------------------------------------
__global__ __launch_bounds__(128) void roth_score_kernel(
    const int* __restrict__ u_idx, const int* __restrict__ v_idx,
    const float* __restrict__ emb_entity,
    const float* __restrict__ bias_head, const float* __restrict__ bias_tail,
    const float* __restrict__ g_ws, float* __restrict__ out)
{
    const int b    = blockIdx.x >> 4;          // 16 tile-groups per b
    const int tg   = blockIdx.x & 15;
    const int wave = threadIdx.x >> 5;
    const int lane = threadIdx.x & 31;
    const int tile = tg * 4 + wave;            // 0..63
    const int j0   = tile * 16;
    const int m    = lane & 15;
    const int c    = lane >> 4;
    const int off  = 2 * c;

    // stage g[b] (2KB) into LDS once per block
    __shared__ float gsh[DD];
    ((float4*)gsh)[threadIdx.x] = ((const float4*)(g_ws + b * DD))[threadIdx.x];
    __syncthreads();

    const int vrow = v_idx[b * MM + j0 + m];
    const float* trow = emb_entity + (long long)vrow * DD;

    v8f acc = {0.f, 0.f, 0.f, 0.f, 0.f, 0.f, 0.f, 0.f};

#pragma unroll 4
    for (int k = 0; k < DD; k += 4) {
        v2f tv = *(const v2f*)(trow + k + off);   // global_load_b64
        v2f gv = *(const v2f*)(gsh  + k + off);   // ds_load_b64
        v2f dv;
        dv.x = gv.x - tv.x;                       // diff = (h+eps) - t
        dv.y = gv.y - tv.y;
        acc = __builtin_amdgcn_wmma_f32_16x16x4_f32(
            false, dv, false, dv, (short)0, acc, false, false);
    }

    // pull diagonal element (j,j): vgpr j&7, lane j (j<8) or j+16 (j>=8)
    float d2 = 0.f;
#pragma unroll
    for (int j = 0; j < 16; ++j) {
        float s = __shfl(acc[j & 7], (j < 8) ? j : (j + 16), 32);
        if (lane == j) d2 = s;
    }

    if (lane < 16) {
        const int j  = j0 + lane;
        const int vr = v_idx[b * MM + j];
        out[b * MM + j] = MARGIN_F - sqrtf(d2)
                        + bias_head[u_idx[b]] + bias_tail[vr];
    }
}

// ---------------------------------------------------------------------------
extern "C" void kernel_launch(void* const* d_in, const int* in_sizes, int n_in,
                              void* d_out, int out_size, void* d_ws, size_t ws_size,
                              hipStream_t stream)
{
    const int*   u_idx               = (const int*)  d_in[0];
    const int*   r_idx               = (const int*)  d_in[1];
    const int*   v_idx               = (const int*)  d_in[2];
    const float* emb_entity          = (const float*)d_in[3];
    const float* relation_rot        = (const float*)d_in[4];
    const float* relation_rot_center = (const float*)d_in[5];
    const float* relation_trans      = (const float*)d_in[6];
    const float* bias_head           = (const float*)d_in[7];
    const float* bias_tail           = (const float*)d_in[8];

    float* out  = (float*)d_out;
    float* g_ws = (float*)d_ws;                // needs 128*512*4 = 256 KB

    roth_head_kernel<<<BB, 256, 0, stream>>>(
        u_idx, r_idx, emb_entity, relation_rot, relation_rot_center,
        relation_trans, g_ws);

    roth_score_kernel<<<BB * 16, 128, 0, stream>>>(
        u_idx, v_idx, emb_entity, bias_head, bias_tail, g_ws, out);
}